// NODEBlock_54460185313871
// MI455X (gfx1250) — compile-verified
//
#include <hip/hip_runtime.h>
#include <math.h>

// NODE block: fused RK4 integrator for dy/dt = tanh(W4·(relu(W2·h1+b2)+h1)+b4),
// h1 = relu(W1·y+b1). 500k independent points, HID=128.
// Compute-bound -> one fully fused kernel; 128x128 matmul on V_WMMA_F32_16X16X4_F32
// (f32 tensor path; f16 WMMA would violate the 1e-5 ODE tolerance).
// Round-3 change: j-outer / n-inner WMMA nesting so A-fragments are short-lived
// -> fits under 256 VGPRs (no s_set_vgpr_msb), single DS base address with
// immediate offsets for all 256 B-fragment loads. Instruction mix unchanged.

typedef __attribute__((ext_vector_type(8))) float v8f;
typedef __attribute__((ext_vector_type(2))) float v2f;

#define HID 128
#define W2S 132           // padded LDS row stride (dwords): lane bank advance 4 -> conflict-free b64
#define SUBSTEPS 8        // RK4 h = 0.25/8 = 1/32 -> global err ~1e-6
#define THREADS 256       // 8 waves * 16 points = 128 points/block
#define PTS_PER_BLOCK 128

// tanh via hw exp2 + rcp: tanh(x) = (e^{2x}-1)/(e^{2x}+1), |err| ~1e-7
__device__ __forceinline__ float fast_tanh(float x) {
    x = fminf(fmaxf(x, -15.0f), 15.0f);           // tanh(±15) == ±1 to 1e-13
    float e = __builtin_amdgcn_exp2f(x * 2.8853900817779268f); // 2*log2(e)
    return (e - 1.0f) * __builtin_amdgcn_rcpf(e + 1.0f);
}

// One dyn() evaluation for this wave's 16 points.
// ys[r][d]  : stage state for rows m = 8*hi + r (replicated across the 16 lanes of a half)
// stg       : LDS staging of all 16 points (needed for A-fragment cross-half access)
// f[r][d]   : output dy/dt, same replication as ys
__device__ __forceinline__ void dyn_eval(
    const float ys[8][3], const float* __restrict__ stg,
    const float* __restrict__ sW2, const float* __restrict__ sW1,
    const float* __restrict__ sB2, const float* __restrict__ sW4,
    float b40, float b41, float b42,
    int lane, int hi, float f[8][3])
{
    const int p = lane & 15;                       // point index for A-layout
    const float yp0 = stg[p * 4 + 0];
    const float yp1 = stg[p * 4 + 1];
    const float yp2 = stg[p * 4 + 2];

    // 8 independent accumulator chains (one per 16-column tile)
    v8f acc[8];
#pragma unroll
    for (int n = 0; n < 8; ++n) acc[n] = (v8f){};

    // Single DS base; every B fragment is base + n*16*W2S*4 + j*16 bytes (imm offsets)
    const float* w2base = &sW2[p * W2S + 2 * hi];

    // ---- layer 1 fragment j computed on the fly, immediately consumed by 8 WMMAs ----
    // fragment j covers k = 4j..4j+3; lane<16 holds (k0,k0+1)=(4j,4j+1), lane>=16 (4j+2,4j+3)
#pragma unroll
    for (int j = 0; j < 32; ++j) {
        const int k0 = 4 * j + 2 * hi;
        const float* w = &sW1[k0 * 4];             // {w0,w1,w2,b1} per row, stride 4
        float t0 = w[0] * yp0 + w[1] * yp1 + w[2] * yp2 + w[3];
        float t1 = w[4] * yp0 + w[5] * yp1 + w[6] * yp2 + w[7];
        v2f aj;
        aj.x = fmaxf(t0, 0.0f);
        aj.y = fmaxf(t1, 0.0f);
#pragma unroll
        for (int n = 0; n < 8; ++n) {
            // B fragment: B[k][col] = W2[col][k], col = n*16+p; (k0,k0+1) -> ds_load_b64
            v2f b = *(const v2f*)(w2base + n * (16 * W2S) + 4 * j);
            acc[n] = __builtin_amdgcn_wmma_f32_16x16x4_f32(
                false, aj, false, b, (short)0, acc[n], false, false);
        }
    }

    // ---- epilogue per 16-column tile; acc[n][r] = (h1 @ W2^T)[m][col], m = r + 8*hi ----
    float part[8][3];
#pragma unroll
    for (int r = 0; r < 8; ++r) { part[r][0] = 0.f; part[r][1] = 0.f; part[r][2] = 0.f; }

#pragma unroll
    for (int n = 0; n < 8; ++n) {
        const int col = n * 16 + p;                // this lane's output column
        const float w1x = sW1[col * 4 + 0], w1y = sW1[col * 4 + 1];
        const float w1z = sW1[col * 4 + 2], w1b = sW1[col * 4 + 3];
        const float bb  = sB2[col];
        const float w40 = sW4[col * 4 + 0], w41 = sW4[col * 4 + 1], w42 = sW4[col * 4 + 2];
#pragma unroll
        for (int r = 0; r < 8; ++r) {
            // residual h1 recomputed in C-layout (3 FMA) instead of cross-lane shuffle
            float h1 = fmaxf(w1x * ys[r][0] + w1y * ys[r][1] + w1z * ys[r][2] + w1b, 0.0f);
            float h2 = fmaxf(acc[n][r] + bb, 0.0f) + h1;
            part[r][0] += h2 * w40;
            part[r][1] += h2 * w41;
            part[r][2] += h2 * w42;
        }
    }

    // ---- layer 4 contraction: butterfly-reduce the 16 lanes of each half ----
#pragma unroll
    for (int r = 0; r < 8; ++r) {
#pragma unroll
        for (int d = 0; d < 3; ++d) {
            float v = part[r][d];
            v += __shfl_xor(v, 1, 32);
            v += __shfl_xor(v, 2, 32);
            v += __shfl_xor(v, 4, 32);
            v += __shfl_xor(v, 8, 32);
            float bd = (d == 0) ? b40 : ((d == 1) ? b41 : b42);
            f[r][d] = fast_tanh(v + bd);
        }
    }
}

__global__ __launch_bounds__(THREADS) void node_rk4_kernel(
    const float* __restrict__ xyz,
    const float* __restrict__ W1, const float* __restrict__ b1,
    const float* __restrict__ W2, const float* __restrict__ b2,
    const float* __restrict__ W4, const float* __restrict__ b4,
    const int* __restrict__ end_time_p, const int* __restrict__ invert_p,
    float* __restrict__ out, int npts, int n_intervals)
{
    __shared__ float sW2[HID * W2S];   // 67.6 KB, padded
    __shared__ float sW1[HID * 4];     // {W1 row, b1}
    __shared__ float sW4[HID * 4];     // {W4 col (3), pad}
    __shared__ float sB2[HID];
    __shared__ float sStage[8 * 16 * 4]; // per-wave 16-point staging

    const int tid  = threadIdx.x;
    const int lane = tid & 31;
    const int wave = tid >> 5;
    const int hi   = lane >> 4;
    const int baseP = blockIdx.x * PTS_PER_BLOCK + wave * 16;

    for (int i = tid; i < HID * HID; i += THREADS)
        sW2[(i >> 7) * W2S + (i & 127)] = W2[i];
    for (int k = tid; k < HID; k += THREADS) {
        sW1[k * 4 + 0] = W1[k * 3 + 0];
        sW1[k * 4 + 1] = W1[k * 3 + 1];
        sW1[k * 4 + 2] = W1[k * 3 + 2];
        sW1[k * 4 + 3] = b1[k];
        sB2[k] = b2[k];
        sW4[k * 4 + 0] = W4[0 * HID + k];
        sW4[k * 4 + 1] = W4[1 * HID + k];
        sW4[k * 4 + 2] = W4[2 * HID + k];
        sW4[k * 4 + 3] = 0.0f;
    }
    __syncthreads();

    const float T   = (float)(*end_time_p);
    const int   inv = *invert_p;
    const float dt  = (inv ? -T : T) / (float)n_intervals;  // autonomous ODE: invert == sign flip
    const float h   = dt / (float)SUBSTEPS;
    const float b40 = b4[0], b41 = b4[1], b42 = b4[2];

    // state y0, replicated per lane-half (rows m = 8*hi + r)
    float y0[8][3];
#pragma unroll
    for (int r = 0; r < 8; ++r) {
        int gp = baseP + 8 * hi + r;
        int cp = gp < npts ? gp : (npts - 1);
#pragma unroll
        for (int d = 0; d < 3; ++d) y0[r][d] = xyz[cp * 3 + d];
    }

    float* stg = &sStage[wave * 64];

    // output row 0 = initial condition (one writer lane per row)
#pragma unroll
    for (int s = 0; s < 8; ++s) {
        if ((lane & 15) == s) {
            int gp = baseP + 8 * hi + s;
            if (gp < npts) {
                float* o = out + (size_t)gp * 3;
                o[0] = y0[s][0]; o[1] = y0[s][1]; o[2] = y0[s][2];
            }
        }
    }

#pragma unroll 1
    for (int iv = 0; iv < n_intervals; ++iv) {
#pragma unroll 1
        for (int ss = 0; ss < SUBSTEPS; ++ss) {
            float ys[8][3], yn[8][3], f[8][3];
#pragma unroll
            for (int r = 0; r < 8; ++r)
#pragma unroll
                for (int d = 0; d < 3; ++d) { ys[r][d] = y0[r][d]; yn[r][d] = y0[r][d]; }

            // RK4 stages are serially dependent: keep this loop ROLLED so the
            // hot sub-step body stays small (I-cache) and under 256 VGPRs.
#pragma unroll 1
            for (int st = 0; st < 4; ++st) {
                const float wgt = (st == 1 || st == 2) ? h * (1.0f / 3.0f) : h * (1.0f / 6.0f);
                const float av  = (st < 2) ? h * 0.5f : h;
                // stage state -> LDS staging (unrolled owner-lane writes, constant slot index)
#pragma unroll
                for (int s = 0; s < 8; ++s) {
                    if ((lane & 15) == s) {
                        int m = 8 * hi + s;
                        stg[m * 4 + 0] = ys[s][0];
                        stg[m * 4 + 1] = ys[s][1];
                        stg[m * 4 + 2] = ys[s][2];
                    }
                }
                __syncthreads();   // fence: staging visible to whole wave before A-layout reads
                dyn_eval(ys, stg, sW2, sW1, sB2, sW4, b40, b41, b42, lane, hi, f);
                // no second barrier: staging is wave-private and same-wave DS ops are in-order
#pragma unroll
                for (int r = 0; r < 8; ++r)
#pragma unroll
                    for (int d = 0; d < 3; ++d) {
                        yn[r][d] += wgt * f[r][d];
                        ys[r][d] = y0[r][d] + av * f[r][d];   // unused after st==3
                    }
            }
#pragma unroll
            for (int r = 0; r < 8; ++r)
#pragma unroll
                for (int d = 0; d < 3; ++d) y0[r][d] = yn[r][d];
        }
        // output row iv+1
#pragma unroll
        for (int s = 0; s < 8; ++s) {
            if ((lane & 15) == s) {
                int gp = baseP + 8 * hi + s;
                if (gp < npts) {
                    float* o = out + ((size_t)(iv + 1) * npts + gp) * 3;
                    o[0] = y0[s][0]; o[1] = y0[s][1]; o[2] = y0[s][2];
                }
            }
        }
    }
}

extern "C" void kernel_launch(void* const* d_in, const int* in_sizes, int n_in,
                              void* d_out, int out_size, void* d_ws, size_t ws_size,
                              hipStream_t stream) {
    (void)n_in; (void)d_ws; (void)ws_size;
    const float* xyz = (const float*)d_in[0];
    const float* W1  = (const float*)d_in[1];
    const float* b1  = (const float*)d_in[2];
    const float* W2  = (const float*)d_in[3];
    const float* b2  = (const float*)d_in[4];
    const float* W4  = (const float*)d_in[5];
    const float* b4  = (const float*)d_in[6];
    const int* end_time = (const int*)d_in[7];
    const int* invert   = (const int*)d_in[9];

    const int npts = in_sizes[0] / 3;
    int n_rows = (npts > 0) ? out_size / (npts * 3) : 2;
    int n_intervals = n_rows - 1;
    if (n_intervals < 1) n_intervals = 1;

    const int blocks = (npts + PTS_PER_BLOCK - 1) / PTS_PER_BLOCK;
    node_rk4_kernel<<<blocks, THREADS, 0, stream>>>(
        xyz, W1, b1, W2, b2, W4, b4, end_time, invert,
        (float*)d_out, npts, n_intervals);
}